// ReconstructedANI_80032420593981
// MI455X (gfx1250) — compile-verified
//
#include <hip/hip_runtime.h>

typedef __attribute__((ext_vector_type(2))) float v2f;
typedef __attribute__((ext_vector_type(8))) float v8f;

#define N_ATOMS 2048
#define NSPEC   7
#define NSHELL  16
#define AEVD    112            // live AEV features (7 species x 16 shells); pad cols are zero
#define RCR_F   5.2f
#define ETA_F   16.0f

// LDS row strides (padded +2 so the 16 rows hit distinct banks on strided b64 loads)
#define SX 114   // 112 + 2
#define S1 258   // 256 + 2
#define S2 194   // 192 + 2
#define S3 162   // 160 + 2

__device__ __constant__ float c_shfr[NSHELL] = {
    0.9f, 1.16f, 1.43f, 1.69f, 1.96f, 2.22f, 2.49f, 2.75f,
    3.02f, 3.28f, 3.55f, 3.81f, 4.08f, 4.34f, 4.61f, 4.87f};

__device__ __forceinline__ float celu01(float x) {
    // celu(x, alpha=0.1) = x>0 ? x : 0.1*(exp(10x)-1)
    return x > 0.f ? x : 0.1f * (__expf(x * 10.f) - 1.f);
}

// ---------------------------------------------------------------------------
// Kernel 1: radial AEV.  One block per atom i; 128 threads sweep neighbors j,
// accumulating the 7x16 species-binned shell features via LDS f32 atomics.
// ---------------------------------------------------------------------------
__global__ void __launch_bounds__(128)
aev_kernel(const int* __restrict__ species,
           const float* __restrict__ coords,
           float* __restrict__ aev) {
    __shared__ float acc[AEVD];
    const int i   = blockIdx.x;
    const int tid = threadIdx.x;
    if (tid < AEVD) acc[tid] = 0.f;
    __syncthreads();

    const float xi = coords[3 * i + 0];
    const float yi = coords[3 * i + 1];
    const float zi = coords[3 * i + 2];

    for (int j = tid; j < N_ATOMS; j += 128) {
        const float dx = xi - coords[3 * j + 0];
        const float dy = yi - coords[3 * j + 1];
        const float dz = zi - coords[3 * j + 2];
        const float sq = dx * dx + dy * dy + dz * dz;
        if (sq > 0.f) {                       // masks the diagonal (and coincident atoms)
            const float d = sqrtf(sq);
            if (d <= RCR_F) {
                const float fc = 0.5f * __cosf(d * (3.14159265358979323846f / RCR_F)) + 0.5f;
                const int base = species[j] * NSHELL;
#pragma unroll
                for (int k = 0; k < NSHELL; ++k) {
                    const float t = d - c_shfr[k];
                    atomicAdd(&acc[base + k], __expf(-ETA_F * t * t) * fc);
                }
            }
        }
    }
    __syncthreads();
    if (tid < AEVD) aev[i * AEVD + tid] = acc[tid];
}

// ---------------------------------------------------------------------------
// One dense layer of the fused MLP:  D(16xNOUT) = celu( src(16xK) * W(KxNOUT) + b )
// computed with V_WMMA_F32_16X16X4_F32.  src lives in LDS, W/b stream from L2,
// dst written back to LDS.  Each of the 8 waves owns N-tiles {w, w+8, ...}.
// A-frag: lane<16 -> {K0,K1} of row M=lane&15 ; lane>=16 -> {K2,K3}.
// B-frag mirrors A.  C/D: vgpr r holds M=r+8*(lane>=16), N=lane&15.
// ---------------------------------------------------------------------------
template <int K, int NOUT, int SSTR, int DSTR>
__device__ __forceinline__ void
mlp_layer(const float* __restrict__ W, const float* __restrict__ Bv,
          const float* __restrict__ src, float* __restrict__ dst,
          int wave, int lane) {
    const int ml = lane & 15;
    const int hi = lane >> 4;              // 0 or 1
    for (int nt = wave; nt < NOUT / 16; nt += 8) {
        const int n = nt * 16 + ml;
        v8f c = {};                        // zero accumulator
#pragma unroll 4
        for (int kb = 0; kb < K; kb += 4) {
            const int ka = kb + 2 * hi;
            const v2f a = *reinterpret_cast<const v2f*>(src + ml * SSTR + ka);
            v2f b;
            b.x = W[(ka + 0) * NOUT + n];
            b.y = W[(ka + 1) * NOUT + n];
            c = __builtin_amdgcn_wmma_f32_16x16x4_f32(
                    /*neg_a=*/false, a, /*neg_b=*/false, b,
                    /*c_mod=*/(short)0, c, /*reuse_a=*/false, /*reuse_b=*/false);
        }
        const float bias = Bv[n];
#pragma unroll
        for (int r = 0; r < 8; ++r) {
            const int m = r + 8 * hi;
            dst[m * DSTR + n] = celu01(c[r] + bias);
        }
    }
}

// ---------------------------------------------------------------------------
// Kernel 2: fused 4-layer per-species MLP.  grid = (128 atom-tiles, 7 species),
// 256 threads (8 waves).  Entire activation chain stays in LDS (~46 KB).
// Layer 3 (160->1) + species routing + tile reduction done by wave 0.
// ---------------------------------------------------------------------------
__global__ void __launch_bounds__(256)
mlp_kernel(const int* __restrict__ species,
           const float* __restrict__ aev,
           const float* __restrict__ W0, const float* __restrict__ B0,
           const float* __restrict__ W1, const float* __restrict__ B1,
           const float* __restrict__ W2, const float* __restrict__ B2,
           const float* __restrict__ W3, const float* __restrict__ B3,
           float* __restrict__ partials) {
    __shared__ __align__(16) float X [16 * SX];
    __shared__ __align__(16) float H1[16 * S1];
    __shared__ __align__(16) float H2[16 * S2];
    __shared__ __align__(16) float H3[16 * S3];

    const int s     = blockIdx.y;          // species (expert) index
    const int tile  = blockIdx.x;          // 16-atom tile
    const int tid   = threadIdx.x;
    const int wave  = tid >> 5;
    const int lane  = tid & 31;
    const int atom0 = tile * 16;

    // Stage the 16x112 AEV tile into LDS (1792 floats / 256 threads = 7 each).
    for (int p = tid; p < 16 * AEVD; p += 256) {
        const int m = p / AEVD, k = p % AEVD;
        X[m * SX + k] = aev[(atom0 + m) * AEVD + k];
    }
    __syncthreads();

    // Layer 0 uses only the first 112 rows of W0 (AEV cols 112..1007 are zero).
    mlp_layer<AEVD, 256, SX, S1>(W0 + (size_t)s * 1008 * 256, B0 + s * 256, X,  H1, wave, lane);
    __syncthreads();
    mlp_layer<256, 192, S1, S2>(W1 + (size_t)s * 256 * 192, B1 + s * 192, H1, H2, wave, lane);
    __syncthreads();
    mlp_layer<192, 160, S2, S3>(W2 + (size_t)s * 192 * 160, B2 + s * 160, H2, H3, wave, lane);
    __syncthreads();

    // Layer 3: e[m] = H3[m,:] . W3[s] + b3[s]; keep only atoms of species s.
    if (wave == 0) {
        float val = 0.f;
        if (lane < 16) {
            const int m = lane;
            const float* __restrict__ w = W3 + s * 160;
            float e = B3[s];
#pragma unroll 8
            for (int k = 0; k < 160; ++k) e += H3[m * S3 + k] * w[k];
            if (species[atom0 + m] == s) val = e;
        }
#pragma unroll
        for (int off = 16; off > 0; off >>= 1) val += __shfl_down(val, off);
        if (lane == 0) partials[blockIdx.y * gridDim.x + blockIdx.x] = val;
    }
}

// ---------------------------------------------------------------------------
// Kernel 3: deterministic final reduction of the 7*128 tile partials.
// ---------------------------------------------------------------------------
__global__ void __launch_bounds__(256)
reduce_kernel(const float* __restrict__ partials, float* __restrict__ out, int n) {
    __shared__ float sm[256];
    float s = 0.f;
    for (int i = threadIdx.x; i < n; i += 256) s += partials[i];
    sm[threadIdx.x] = s;
    __syncthreads();
    for (int w = 128; w > 0; w >>= 1) {
        if (threadIdx.x < w) sm[threadIdx.x] += sm[threadIdx.x + w];
        __syncthreads();
    }
    if (threadIdx.x == 0) out[0] = sm[0];
}

// ---------------------------------------------------------------------------
extern "C" void kernel_launch(void* const* d_in, const int* in_sizes, int n_in,
                              void* d_out, int out_size, void* d_ws, size_t ws_size,
                              hipStream_t stream) {
    (void)in_sizes; (void)n_in; (void)out_size; (void)ws_size;

    const int*   species = (const int*)  d_in[0];   // [1,2048] int32
    const float* coords  = (const float*)d_in[1];   // [1,2048,3]
    const float* W0 = (const float*)d_in[2];        // [7,1008,256]
    const float* B0 = (const float*)d_in[3];        // [7,256]
    const float* W1 = (const float*)d_in[4];        // [7,256,192]
    const float* B1 = (const float*)d_in[5];        // [7,192]
    const float* W2 = (const float*)d_in[6];        // [7,192,160]
    const float* B2 = (const float*)d_in[7];        // [7,160]
    const float* W3 = (const float*)d_in[8];        // [7,160,1]
    const float* B3 = (const float*)d_in[9];        // [7,1]

    float* aev      = (float*)d_ws;                 // 2048*112 f32  (~0.92 MB)
    float* partials = aev + (size_t)N_ATOMS * AEVD; // 7*128 f32

    aev_kernel<<<N_ATOMS, 128, 0, stream>>>(species, coords, aev);
    mlp_kernel<<<dim3(N_ATOMS / 16, NSPEC), 256, 0, stream>>>(
        species, aev, W0, B0, W1, B1, W2, B2, W3, B3, partials);
    reduce_kernel<<<1, 256, 0, stream>>>(partials, (float*)d_out, NSPEC * (N_ATOMS / 16));
}